// NURBSSurfaces_20100446945896
// MI455X (gfx1250) — compile-verified
//
#include <hip/hip_runtime.h>
#include <cstdint>

#define NCP      32
#define DEG      3
#define NSPAN    29          // n_cp - p
#define NSLICE   16          // S * F
#define NTHREADS 256
#define NBLK_X   64
#define CPFLOATS (NCP * NCP * 3)   // 3072 floats = 12288 bytes per slice

// clamped cubic knot vector for 32 control points:
// kv[k] = clamp((k-3)/29, 0, 1), k in [0, 35]
__device__ __forceinline__ float knotv(int k) {
    float t = (float)(k - DEG) * (1.0f / (float)NSPAN);
    t = t < 0.0f ? 0.0f : t;
    return t > 1.0f ? 1.0f : t;
}

// Cox–de Boor basis + first derivative, p = 3, fully unrolled.
__device__ __forceinline__ void basis_deriv(float u, int span, float N0[4], float N1[4]) {
    float left[DEG + 1], right[DEG + 1];
    float ndu[DEG + 1][DEG + 1];
    ndu[0][0] = 1.0f;
#pragma unroll
    for (int j = 1; j <= DEG; ++j) {
        left[j]  = u - knotv(span - j + 1);
        right[j] = knotv(span + j) - u;
        float saved = 0.0f;
#pragma unroll
        for (int r = 0; r < j; ++r) {
            ndu[j][r] = right[r + 1] + left[j - r];              // knot difference
            float tmp = ndu[r][j - 1] * __builtin_amdgcn_rcpf(ndu[j][r]);
            ndu[r][j] = fmaf(right[r + 1], tmp, saved);
            saved = left[j - r] * tmp;
        }
        ndu[j][j] = saved;
    }
    N0[0] = ndu[0][3]; N0[1] = ndu[1][3]; N0[2] = ndu[2][3]; N0[3] = ndu[3][3];
    // ders[1][r] = ndu[r-1][2]/ndu[3][r-1] - ndu[r][2]/ndu[3][r]   (terms dropped OOR)
    float d0 = ndu[0][2] * __builtin_amdgcn_rcpf(ndu[3][0]);
    float d1 = ndu[1][2] * __builtin_amdgcn_rcpf(ndu[3][1]);
    float d2 = ndu[2][2] * __builtin_amdgcn_rcpf(ndu[3][2]);
    N1[0] = -3.0f * d0;
    N1[1] =  3.0f * (d0 - d1);
    N1[2] =  3.0f * (d1 - d2);
    N1[3] =  3.0f * d2;
}

__global__ __launch_bounds__(NTHREADS)
void nurbs_surface_eval(const float* __restrict__ eval_pts,  // (16, E, 2)
                        const float* __restrict__ ctrl_pts,  // (16, 32, 32, 3)
                        float* __restrict__ out,             // points4 (16,E,4) ++ normals4 (16,E,4)
                        int E) {
    __shared__ __align__(16) float cp[CPFLOATS];   // 12 KB control grid for this slice

    const int slice = blockIdx.y;
    const int tid   = threadIdx.x;

    // ---- Stage this slice's 12 KB control-point grid into LDS (async DMA path) ----
#if defined(__HIP_DEVICE_COMPILE__)
    {
        // raw 32-bit LDS byte offset (generic LDS ptr low 32 bits == ds offset on gfx12xx)
        unsigned lbase = (unsigned)(uintptr_t)(void*)cp;
        unsigned long long gbase =
            (unsigned long long)(ctrl_pts + (size_t)slice * CPFLOATS);
        // 12288 B = 768 x b128 transfers; 256 lanes x 3 each, all lanes active here
#pragma unroll
        for (int k = 0; k < 3; ++k) {
            unsigned byteoff = (unsigned)((tid + k * NTHREADS) * 16);
            unsigned laddr = lbase + byteoff;
            unsigned long long gaddr = gbase + byteoff;
            asm volatile("global_load_async_to_lds_b128 %0, %1, off"
                         :: "v"(laddr), "v"(gaddr) : "memory");
        }
        asm volatile("s_wait_asynccnt 0x0" ::: "memory");
    }
#else
    for (int k = tid; k < CPFLOATS; k += NTHREADS)
        cp[k] = ctrl_pts[(size_t)slice * CPFLOATS + k];
#endif
    __syncthreads();

    const float2* ep = (const float2*)eval_pts + (size_t)slice * E;
    float4* outP = (float4*)out + (size_t)slice * E;
    float4* outN = (float4*)out + (size_t)NSLICE * E + (size_t)slice * E;

    const int stride = NBLK_X * NTHREADS;
    for (int e = blockIdx.x * NTHREADS + tid; e < E; e += stride) {
        if (e + stride < E)
            __builtin_prefetch(&ep[e + stride], 0, 0);   // global_prefetch_b8

        float2 uv = ep[e];
        float u = uv.x, v = uv.y;

        int su = (int)(u * (float)NSPAN) + DEG;
        su = su < DEG ? DEG : (su > NCP - 1 ? NCP - 1 : su);
        int sv = (int)(v * (float)NSPAN) + DEG;
        sv = sv < DEG ? DEG : (sv > NCP - 1 ? NCP - 1 : sv);

        float Nu0[4], Nu1[4], Nv0[4], Nv1[4];
        basis_deriv(u, su, Nu0, Nu1);
        basis_deriv(v, sv, Nv0, Nv1);

        const int iu0 = su - DEG, iv0 = sv - DEG;
        float px = 0.f, py = 0.f, pz = 0.f;
        float dux = 0.f, duy = 0.f, duz = 0.f;
        float dvx = 0.f, dvy = 0.f, dvz = 0.f;

        // Separable contraction: first over j (v-direction), then over i (u-direction).
        //   A_i = sum_j g_ij * Nv0[j],  B_i = sum_j g_ij * Nv1[j]
        //   pts = sum_i Nu0[i]*A_i,  du = sum_i Nu1[i]*A_i,  dv = sum_i Nu0[i]*B_i
#pragma unroll
        for (int i = 0; i < 4; ++i) {
            const int rowbase = ((iu0 + i) * NCP + iv0) * 3;
            float ax = 0.f, ay = 0.f, az = 0.f;
            float bx = 0.f, by = 0.f, bz = 0.f;
#pragma unroll
            for (int j = 0; j < 4; ++j) {
                float gx = cp[rowbase + 3 * j + 0];
                float gy = cp[rowbase + 3 * j + 1];
                float gz = cp[rowbase + 3 * j + 2];
                float w0 = Nv0[j], w1 = Nv1[j];
                ax = fmaf(gx, w0, ax);  ay = fmaf(gy, w0, ay);  az = fmaf(gz, w0, az);
                bx = fmaf(gx, w1, bx);  by = fmaf(gy, w1, by);  bz = fmaf(gz, w1, bz);
            }
            const float cu0 = Nu0[i], cu1 = Nu1[i];
            px  = fmaf(ax, cu0, px );  py  = fmaf(ay, cu0, py );  pz  = fmaf(az, cu0, pz );
            dux = fmaf(ax, cu1, dux);  duy = fmaf(ay, cu1, duy);  duz = fmaf(az, cu1, duz);
            dvx = fmaf(bx, cu0, dvx);  dvy = fmaf(by, cu0, dvy);  dvz = fmaf(bz, cu0, dvz);
        }

        // normal = normalize(du x dv)
        float nx = fmaf(duy, dvz, -duz * dvy);
        float ny = fmaf(duz, dvx, -dux * dvz);
        float nz = fmaf(dux, dvy, -duy * dvx);
        float len = sqrtf(fmaf(nx, nx, fmaf(ny, ny, nz * nz)));
        float inv = __builtin_amdgcn_rcpf(len + 1e-12f);

        outP[e] = make_float4(px, py, pz, 1.0f);
        outN[e] = make_float4(nx * inv, ny * inv, nz * inv, 0.0f);
    }
}

extern "C" void kernel_launch(void* const* d_in, const int* in_sizes, int n_in,
                              void* d_out, int out_size, void* d_ws, size_t ws_size,
                              hipStream_t stream) {
    const float* eval_pts = (const float*)d_in[0];   // (4,4,E,2) f32
    const float* ctrl_pts = (const float*)d_in[1];   // (4,4,32,32,3) f32
    float* out = (float*)d_out;                      // points4 ++ normals4

    const int E = in_sizes[0] / (NSLICE * 2);        // 65536 for the reference shapes
    dim3 grid(NBLK_X, NSLICE, 1);
    nurbs_surface_eval<<<grid, NTHREADS, 0, stream>>>(eval_pts, ctrl_pts, out, E);
}